// GNN_27041114095623
// MI455X (gfx1250) — compile-verified
//
#include <hip/hip_runtime.h>
#include <hip/hip_bf16.h>
#include <math.h>

typedef __attribute__((ext_vector_type(2))) float v2f;
typedef __attribute__((ext_vector_type(8))) float v8f;

#define TPB 256

// ---- monotone float<->uint key for atomicMax on floats ----
__device__ __forceinline__ unsigned f2key(float f) {
  unsigned u = __float_as_uint(f);
  return (u & 0x80000000u) ? ~u : (u | 0x80000000u);
}
__device__ __forceinline__ float key2f(unsigned k) {
  return __uint_as_float((k & 0x80000000u) ? (k & 0x7FFFFFFFu) : ~k);
}

// deg[i] = 1 (self loop); zero the scalar cells (maxkey, sumexp, colsum[16])
__global__ void k_init(float* __restrict__ deg, float* __restrict__ scal, int N) {
  int i = blockIdx.x * blockDim.x + threadIdx.x;
  if (i < N) deg[i] = 1.0f;
  if (i < 32) scal[i] = 0.0f;
}

__global__ void k_deg(const int* __restrict__ dst, float* __restrict__ deg, int E) {
  int e = blockIdx.x * blockDim.x + threadIdx.x;
  if (e < E) atomicAdd(&deg[dst[e]], 1.0f);
}

__global__ void k_dinv(float* __restrict__ d, int N) {
  int i = blockIdx.x * blockDim.x + threadIdx.x;
  if (i < N) { float v = d[i]; d[i] = (v > 0.0f) ? rsqrtf(v) : 0.0f; }
}

// out[n x 16] = A[n x K] @ W[K x 16], fp32 WMMA 16x16x4, one wave per 16-row tile.
// n must be a multiple of 16 (true here: 100000).
__global__ void k_gemm_wmma(const float* __restrict__ A, const float* __restrict__ W,
                            float* __restrict__ out, int n, int K) {
  int wave = (int)((blockIdx.x * blockDim.x + threadIdx.x) >> 5);
  int lane = threadIdx.x & 31;
  int row0 = wave << 4;
  if (row0 >= n) return;                  // wave-uniform: EXEC stays all-ones
  int half = lane >> 4;                   // 0: K pair {0,1}; 1: K pair {2,3}
  int l = lane & 15;                      // M index for A, N index for B/D
  const float* arow = A + (size_t)(row0 + l) * K;
  v8f acc = {};
  for (int k = 0; k < K; k += 4) {
    int ka = k + half * 2;
    v2f a, b;
    a.x = arow[ka];
    a.y = arow[ka + 1];
    b.x = W[(size_t)ka * 16 + l];
    b.y = W[(size_t)(ka + 1) * 16 + l];
    acc = __builtin_amdgcn_wmma_f32_16x16x4_f32(
        /*neg_a=*/false, a, /*neg_b=*/false, b,
        /*c_mod=*/(short)0, acc, /*reuse_a=*/false, /*reuse_b=*/false);
  }
  // D layout: VGPR v, lanes 0-15 -> M=v, lanes 16-31 -> M=v+8; N=l
  float* o = out + (size_t)(row0 + half * 8) * 16 + l;
#pragma unroll
  for (int v = 0; v < 8; ++v) o[(size_t)v * 16] = acc[v];
}

// g = dinv[node] * h ; acc = g  (acc init covers the self-loop term)
__global__ void k_prescale(const float* __restrict__ h, const float* __restrict__ dinv,
                           float* __restrict__ g, float* __restrict__ acc, long long M) {
  long long i = (long long)blockIdx.x * blockDim.x + threadIdx.x;
  if (i < M) { float v = dinv[i >> 4] * h[i]; g[i] = v; acc[i] = v; }
}

// one lane per (edge, feature): acc[dst*16+j] += g[src*16+j]
__global__ void k_scatter16(const int* __restrict__ src, const int* __restrict__ dst,
                            const float* __restrict__ g, float* __restrict__ acc,
                            long long EF) {
  long long i = (long long)blockIdx.x * blockDim.x + threadIdx.x;
  if (i >= EF) return;
  int e = (int)(i >> 4);
  int j = (int)(i & 15);
  int s = src[e], d = dst[e];
  atomicAdd(&acc[(size_t)d * 16 + j], g[(size_t)s * 16 + j]);
}

// out = relu(dinv[node]*acc + b[feat])
__global__ void k_finish_relu(const float* __restrict__ acc, const float* __restrict__ dinv,
                              const float* __restrict__ b, float* __restrict__ out,
                              long long M) {
  long long i = (long long)blockIdx.x * blockDim.x + threadIdx.x;
  if (i < M) {
    float v = dinv[i >> 4] * acc[i] + b[i & 15];
    out[i] = fmaxf(v, 0.0f);
  }
}

// layer-3 node stage: cpre = h2[v].W3 ; g3 = dinv*cpre ; acc3 = g3
__global__ void k_l3_node(const float* __restrict__ h2, const float* __restrict__ W3,
                          const float* __restrict__ dinv, float* __restrict__ g3,
                          float* __restrict__ acc3, int N) {
  int v = blockIdx.x * blockDim.x + threadIdx.x;
  if (v >= N) return;
  float s = 0.0f;
#pragma unroll
  for (int j = 0; j < 16; ++j) s += h2[(size_t)v * 16 + j] * W3[j];
  float gv = dinv[v] * s;
  g3[v] = gv;
  acc3[v] = gv;
}

__global__ void k_scatter1(const int* __restrict__ src, const int* __restrict__ dst,
                           const float* __restrict__ g3, float* __restrict__ acc3, int E) {
  int e = blockIdx.x * blockDim.x + threadIdx.x;
  if (e < E) atomicAdd(&acc3[dst[e]], g3[src[e]]);
}

// c = dinv*acc3 + b3 ; block-max -> atomicMax(key)
__global__ void k_c_max(const float* __restrict__ acc3, const float* __restrict__ dinv,
                        const float* __restrict__ b3, float* __restrict__ c,
                        unsigned* __restrict__ maxkey, int N) {
  __shared__ unsigned sk[TPB];
  int i = blockIdx.x * blockDim.x + threadIdx.x;
  unsigned key = 0u;
  if (i < N) {
    float cv = dinv[i] * acc3[i] + b3[0];
    c[i] = cv;
    key = f2key(cv);
  }
  sk[threadIdx.x] = key;
  __syncthreads();
  for (int off = TPB / 2; off > 0; off >>= 1) {
    if (threadIdx.x < off) {
      unsigned o = sk[threadIdx.x + off];
      if (o > sk[threadIdx.x]) sk[threadIdx.x] = o;
    }
    __syncthreads();
  }
  if (threadIdx.x == 0) atomicMax(maxkey, sk[0]);
}

__global__ void k_sumexp(const float* __restrict__ c, const unsigned* __restrict__ maxkey,
                         float* __restrict__ sum, int N) {
  __shared__ float ss[TPB];
  int i = blockIdx.x * blockDim.x + threadIdx.x;
  float m = key2f(*maxkey);
  ss[threadIdx.x] = (i < N) ? expf(c[i] - m) : 0.0f;
  __syncthreads();
  for (int off = TPB / 2; off > 0; off >>= 1) {
    if (threadIdx.x < off) ss[threadIdx.x] += ss[threadIdx.x + off];
    __syncthreads();
  }
  if (threadIdx.x == 0) atomicAdd(sum, ss[0]);
}

__global__ void k_choice(const float* __restrict__ c, const unsigned* __restrict__ maxkey,
                         const float* __restrict__ sum, float* __restrict__ out, int N) {
  int i = blockIdx.x * blockDim.x + threadIdx.x;
  if (i < N) out[i] = expf(c[i] - key2f(*maxkey)) / (*sum);
}

// column sums of h2 (for global mean pool)
__global__ void k_colsum(const float* __restrict__ h2, float* __restrict__ colsum, int N) {
  __shared__ float ss[16];
  if (threadIdx.x < 16) ss[threadIdx.x] = 0.0f;
  __syncthreads();
  int i = blockIdx.x * blockDim.x + threadIdx.x;
  if (i < N) {
#pragma unroll
    for (int j = 0; j < 16; ++j) atomicAdd(&ss[j], h2[(size_t)i * 16 + j]);
  }
  __syncthreads();
  if (threadIdx.x < 16) atomicAdd(&colsum[threadIdx.x], ss[threadIdx.x]);
}

__global__ void k_value(const float* __restrict__ colsum, const float* __restrict__ Wv,
                        const float* __restrict__ bv, float* __restrict__ out, int N) {
  if (blockIdx.x == 0 && threadIdx.x == 0) {
    float s = 0.0f;
    float invN = 1.0f / (float)N;
#pragma unroll
    for (int j = 0; j < 16; ++j) s += (colsum[j] * invN) * Wv[j];
    out[N] = s + bv[0];
  }
}

extern "C" void kernel_launch(void* const* d_in, const int* in_sizes, int n_in,
                              void* d_out, int out_size, void* d_ws, size_t ws_size,
                              hipStream_t stream) {
  const float* x  = (const float*)d_in[0];
  const int*   ei = (const int*)d_in[1];
  const float* W1 = (const float*)d_in[2];
  const float* b1 = (const float*)d_in[3];
  const float* W2 = (const float*)d_in[4];
  const float* b2 = (const float*)d_in[5];
  const float* W3 = (const float*)d_in[6];
  const float* b3 = (const float*)d_in[7];
  const float* Wv = (const float*)d_in[8];
  const float* bv = (const float*)d_in[9];

  const int N = in_sizes[0] / 128;
  const int E = in_sizes[1] / 2;
  const int* src = ei;
  const int* dst = ei + E;

  // ws layout (floats): dinv[N] | P1[16N] | P2[16N] | P3[16N] | scal[32]
  float* ws   = (float*)d_ws;
  float* dinv = ws;
  float* P1   = dinv + N;
  float* P2   = P1 + (size_t)16 * N;
  float* P3   = P2 + (size_t)16 * N;
  float* scal = P3 + (size_t)16 * N;
  unsigned* maxkey = (unsigned*)&scal[0];
  float* sumexp    = &scal[1];
  float* colsum    = &scal[2];

  float* out = (float*)d_out;
  const long long M  = (long long)N * 16;
  const long long EF = (long long)E * 16;
  auto nb = [](long long n) { return (int)((n + TPB - 1) / TPB); };
  const int gemm_threads = ((N + 15) / 16) * 32;  // one wave per 16-row tile

  // degree / normalization
  k_init<<<nb(N), TPB, 0, stream>>>(dinv, scal, N);
  k_deg<<<nb(E), TPB, 0, stream>>>(dst, dinv, E);
  k_dinv<<<nb(N), TPB, 0, stream>>>(dinv, N);

  // layer 1: h1 = relu(Agg(x @ W1) + b1)   (h1 -> P1)
  k_gemm_wmma<<<nb(gemm_threads), TPB, 0, stream>>>(x, W1, P1, N, 128);
  k_prescale<<<nb(M), TPB, 0, stream>>>(P1, dinv, P2, P3, M);
  k_scatter16<<<nb(EF), TPB, 0, stream>>>(src, dst, P2, P3, EF);
  k_finish_relu<<<nb(M), TPB, 0, stream>>>(P3, dinv, b1, P1, M);

  // layer 2: h2 = relu(Agg(h1 @ W2) + b2)  (h2 -> P2)
  k_gemm_wmma<<<nb(gemm_threads), TPB, 0, stream>>>(P1, W2, P2, N, 16);
  k_prescale<<<nb(M), TPB, 0, stream>>>(P2, dinv, P3, P1, M);
  k_scatter16<<<nb(EF), TPB, 0, stream>>>(src, dst, P3, P1, EF);
  k_finish_relu<<<nb(M), TPB, 0, stream>>>(P1, dinv, b2, P2, M);

  // layer 3 (scalar channel) + softmax over nodes
  float* g3   = P3;
  float* acc3 = P3 + N;
  float* c    = P3 + (size_t)2 * N;
  k_l3_node<<<nb(N), TPB, 0, stream>>>(P2, W3, dinv, g3, acc3, N);
  k_scatter1<<<nb(E), TPB, 0, stream>>>(src, dst, g3, acc3, E);
  k_c_max<<<nb(N), TPB, 0, stream>>>(acc3, dinv, b3, c, maxkey, N);
  k_sumexp<<<nb(N), TPB, 0, stream>>>(c, maxkey, sumexp, N);
  k_choice<<<nb(N), TPB, 0, stream>>>(c, maxkey, sumexp, out, N);

  // value head: mean-pool h2 then dot with Wv
  k_colsum<<<nb(N), TPB, 0, stream>>>(P2, colsum, N);
  k_value<<<1, 1, 0, stream>>>(colsum, Wv, bv, out, N);
}